// CumulantSOAP_CV_82111184765535
// MI455X (gfx1250) — compile-verified
//
#include <hip/hip_runtime.h>
#include <hip/hip_bf16.h>

// CumulantSOAP on MI455X (gfx1250, wave32).
//
// Phase 1 (bandwidth-bound, ~17.6us floor @ 23.3 TB/s):
//   stream X (100000 x 1024 f32, 409.6 MB) once with non-temporal 128-bit
//   loads, accumulate raw power sums S1..S5 per column. Each block covers all
//   1024 columns (256 threads x float4), handles a row chunk, writes 5120 f32
//   partials to d_ws. X is single-use and > 192 MB L2, hence TH=NT.
// Phase 2 (negligible size):
//   single block reduces partials, converts raw->central moments, forms
//   v = cumulants - mu, then computes v @ W (K=5120, Nout=8) with
//   V_WMMA_F32_16X16X4_F32 (f32 WMMA == reference precision), K split
//   across the 8 waves, cross-wave reduce in LDS. Operands are built with
//   arithmetic lane masks (x*0 == 0 for finite x) so every lane consumes
//   every load -> no exec-mask ladders, EXEC stays all-ones through WMMA.

#define PCOLS 1024
#define NC 5
#define CUMLEN (PCOLS * NC)   // 5120

typedef __attribute__((ext_vector_type(2))) float v2f;
typedef __attribute__((ext_vector_type(4))) float v4f;
typedef __attribute__((ext_vector_type(8))) float v8f;

__global__ __launch_bounds__(256)
void cumulant_partials_kernel(const float* __restrict__ X,
                              float* __restrict__ ws,
                              int N, int rows_per_block) {
    const int b = blockIdx.x;
    const int t = threadIdx.x;          // 0..255
    const int col4 = t * 4;             // 4 consecutive columns per thread

    int r0 = b * rows_per_block;
    int r1 = r0 + rows_per_block;
    if (r1 > N) r1 = N;

    float s[4][NC];
#pragma unroll
    for (int c = 0; c < 4; ++c)
#pragma unroll
        for (int k = 0; k < NC; ++k) s[c][k] = 0.0f;

    const float* __restrict__ base = X + (size_t)r0 * PCOLS + col4;
#pragma unroll 4
    for (int r = r0; r < r1; ++r) {
        // Non-temporal 128-bit load: X is streamed exactly once (410 MB > L2).
        const v4f x4 = __builtin_nontemporal_load((const v4f*)base);
        base += PCOLS;
#pragma unroll
        for (int c = 0; c < 4; ++c) {
            const float x  = x4[c];
            const float x2 = x * x;
            const float x3 = x2 * x;
            s[c][0] += x;
            s[c][1] += x2;
            s[c][2] += x3;
            s[c][3] += x2 * x2;
            s[c][4] += x3 * x2;
        }
    }

    float* __restrict__ out = ws + (size_t)b * CUMLEN + col4 * NC;
#pragma unroll
    for (int c = 0; c < 4; ++c)
#pragma unroll
        for (int k = 0; k < NC; ++k)
            out[c * NC + k] = s[c][k];
}

__global__ __launch_bounds__(256)
void cumulant_finalize_kernel(const float* __restrict__ ws,
                              const float* __restrict__ mu_in,
                              const float* __restrict__ W,
                              float* __restrict__ out,
                              int R, int N) {
    __shared__ float sums[CUMLEN];   // 20 KB
    __shared__ float vbuf[CUMLEN];   // 20 KB
    __shared__ float red[8 * 16];

    const int t = threadIdx.x;       // 0..255

    // ---- Stage A: reduce R block-partials (coalesced over t) ----
#pragma unroll
    for (int i = 0; i < CUMLEN / 256; ++i) {      // 20 iters
        const int idx = i * 256 + t;
        float acc = 0.0f;
        for (int b = 0; b < R; ++b)
            acc += ws[(size_t)b * CUMLEN + idx];
        sums[idx] = acc;
    }
    __syncthreads();

    // ---- Stage B: raw moments -> cumulants, subtract mu ----
    const float invN = 1.0f / (float)N;
#pragma unroll
    for (int cc = 0; cc < 4; ++cc) {
        const int col  = cc * 256 + t;
        const int base = col * NC;
        const float M1 = sums[base + 0] * invN;
        const float M2 = sums[base + 1] * invN;
        const float M3 = sums[base + 2] * invN;
        const float M4 = sums[base + 3] * invN;
        const float M5 = sums[base + 4] * invN;
        const float mu  = M1;
        const float mu2 = mu * mu;
        const float mu3 = mu2 * mu;
        const float m2 = M2 - mu2;
        const float m3 = M3 - 3.0f * mu * M2 + 2.0f * mu3;
        const float m5 = M5 - 5.0f * mu * M4 + 10.0f * mu2 * M3
                       - 10.0f * mu3 * M2 + 4.0f * mu2 * mu3;
        const float c0 = mu;
        const float c1 = 0.0f;               // mean(centered): exactly 0 analytically
        const float c2 = m2;
        const float c3 = m3 - 3.0f * m2 * m2;
        const float c4 = m5 - 10.0f * m2 * m3;
        vbuf[base + 0] = c0 - mu_in[base + 0];
        vbuf[base + 1] = c1 - mu_in[base + 1];
        vbuf[base + 2] = c2 - mu_in[base + 2];
        vbuf[base + 3] = c3 - mu_in[base + 3];
        vbuf[base + 4] = c4 - mu_in[base + 4];
    }
    __syncthreads();

    // ---- Stage C: v @ W via V_WMMA_F32_16X16X4_F32 ----
    // K = 5120 split across 8 waves (640 each = 160 WMMAs/wave).
    // A (16x4 f32, row 0 = v chunk): vgpr = k&1, lane-half selects k>=2.
    // B (4x16 f32, N = lane%16):     vgpr = k&1, lane-half selects k>=2.
    // Lane masking is arithmetic (0/1 multipliers): every lane consumes every
    // load, so the compiler cannot sink loads into divergent regions -> the
    // loop is branch-free and EXEC stays all-ones (WMMA requirement).
    const int w    = t >> 5;         // wave id 0..7
    const int lane = t & 31;
    const int n    = lane & 15;
    const bool hi  = lane >= 16;
    const float m0   = (lane == 0)  ? 1.0f : 0.0f;
    const float m16  = (lane == 16) ? 1.0f : 0.0f;
    const float nmsk = (n < 8)      ? 1.0f : 0.0f;
    const int   ncl  = (n < 8) ? n : 7;   // clamped W column (kept in-bounds)

    v8f acc = {0.0f, 0.0f, 0.0f, 0.0f, 0.0f, 0.0f, 0.0f, 0.0f};
    const int kbeg = w * (CUMLEN / 8);
    const int kend = kbeg + (CUMLEN / 8);
    for (int j = kbeg; j < kend; j += 4) {
        // All lanes read all 4 v values (uniform address -> broadcast).
        const float va0 = vbuf[j + 0];
        const float va1 = vbuf[j + 1];
        const float va2 = vbuf[j + 2];
        const float va3 = vbuf[j + 3];
        v2f a;
        a[0] = va0 * m0 + va2 * m16;   // exact: all values finite
        a[1] = va1 * m0 + va3 * m16;

        const int k0 = hi ? (j + 2) : (j + 0);
        const int k1 = hi ? (j + 3) : (j + 1);
        const float wb0 = W[k0 * 8 + ncl];
        const float wb1 = W[k1 * 8 + ncl];
        v2f b;
        b[0] = wb0 * nmsk;
        b[1] = wb1 * nmsk;

        acc = __builtin_amdgcn_wmma_f32_16x16x4_f32(
            /*neg_a=*/false, a, /*neg_b=*/false, b,
            /*c_mod=*/(short)0, acc, /*reuse_a=*/false, /*reuse_b=*/false);
    }

    // D row M=0 lives in acc[0] on lanes 0..15 (N = lane).
    if (lane < 16) red[w * 16 + lane] = acc[0];
    __syncthreads();

    if (t < 8) {
        float o = 0.0f;
#pragma unroll
        for (int ww = 0; ww < 8; ++ww) o += red[ww * 16 + t];
        out[t] = o;
    }
}

extern "C" void kernel_launch(void* const* d_in, const int* in_sizes, int n_in,
                              void* d_out, int out_size, void* d_ws, size_t ws_size,
                              hipStream_t stream) {
    const float* X  = (const float*)d_in[0];
    const float* mu = (const float*)d_in[1];
    const float* W  = (const float*)d_in[2];
    float* out      = (float*)d_out;
    float* ws       = (float*)d_ws;

    const int N = in_sizes[0] / PCOLS;   // 100000

    // Row-chunk count: as parallel as scratch allows, capped at 512.
    size_t cap = ws_size / (CUMLEN * sizeof(float));
    int R = (cap < 512) ? (int)cap : 512;
    if (R < 1) R = 1;
    const int rows_per_block = (N + R - 1) / R;

    cumulant_partials_kernel<<<R, 256, 0, stream>>>(X, ws, N, rows_per_block);
    cumulant_finalize_kernel<<<1, 256, 0, stream>>>(ws, mu, W, out, R, N);
}